// Gaussian_65962107732122
// MI455X (gfx1250) — compile-verified
//
#include <hip/hip_runtime.h>
#include <stdint.h>

typedef float    v4f   __attribute__((ext_vector_type(4)));
typedef float    v2f   __attribute__((ext_vector_type(2)));
typedef uint32_t u32x4 __attribute__((ext_vector_type(4)));
typedef int      i32x4 __attribute__((ext_vector_type(4)));
typedef int      i32x8 __attribute__((ext_vector_type(8)));

#define B_   512
#define C_   64
#define G_   16
#define F_   128
#define CG_  1024   // C*G outputs per b
#define SEGS 16     // blocks per b  -> 64 outputs per block
#define WAVES 8

// out[b, cg] = exp(-0.5 * sum_f ((x[b,f]-mean)^2 / var^2)),
// rep laid out as [b, cg, f, {mean,var}] contiguous f32 pairs (256 f32 per cg).
//
// Each wave32 handles TWO outputs per iteration: lanes 0-15 -> cg, lanes
// 16-31 -> cg+1. Lane 'sl' (=lane&15) owns, for each k in 0..3, the 16-byte
// chunk at float offset cg*256 + k*64 + sl*4 (= features k*32+sl*2, +1), so
// every load instruction covers two contiguous 256 B segments. The reduction
// is a 4-step xor-butterfly confined to each 16-lane half.
__global__ __launch_bounds__(256) void gaussian_quad_kernel(
    const float* __restrict__ x,
    const float* __restrict__ rep,
    float* __restrict__ out)
{
    __shared__ __align__(16) float xs[F_];

    const int b    = blockIdx.x >> 4;
    const int seg  = blockIdx.x & 15;
    const int tid  = threadIdx.x;
    const int wave = tid >> 5;
    const int lane = tid & 31;
    const int sub  = lane >> 4;   // which output of the pair
    const int sl   = lane & 15;   // lane within 16-lane sub-group

    // ---- Stage x[b, 0:128] (512 B) into LDS via the Tensor Data Mover ----
#if defined(__has_builtin)
#if __has_builtin(__builtin_amdgcn_tensor_load_to_lds) && __has_builtin(__builtin_amdgcn_s_wait_tensorcnt)
#define XS_TDM_STAGED 1
    if (tid < 32) {  // wave 0 issues the TDM op
        const uint64_t ga  = (uint64_t)(uintptr_t)(x + (size_t)b * F_);
        const uint32_t lds = (uint32_t)(uintptr_t)(void*)&xs[0]; // low 32b = LDS offset
        // D# group0: count=1 | lds_addr | global_addr[56:0] | type=2 @127:126
        u32x4 g0 = { 1u, lds, (uint32_t)ga, (uint32_t)(ga >> 32) | (2u << 30) };
        // D# group1: wg_mask=0, data_size=2(4B) @17:16, tensor_dim0=128 @79:48,
        //            tensor_dim1=1 @111:80, tile_dim0=128 @127:112,
        //            tile_dim1=1 @143:128, tensor_dim0_stride=128 @207:160
        i32x8 g1 = { (int)(2u << 16),
                     (int)(128u << 16),
                     (int)(1u << 16),
                     (int)(128u << 16),
                     1,
                     128,
                     0, 0 };
        i32x4 g2 = { 0, 0, 0, 0 };   // unused (<=2D tensor)
        i32x4 g3 = { 0, 0, 0, 0 };
#if __clang_major__ >= 23
        i32x8 g4 = { 0, 0, 0, 0, 0, 0, 0, 0 };
        __builtin_amdgcn_tensor_load_to_lds(g0, g1, g2, g3, g4, 0);
#else
        __builtin_amdgcn_tensor_load_to_lds(g0, g1, g2, g3, 0);
#endif
        __builtin_amdgcn_s_wait_tensorcnt(0);
    }
#endif
#endif
#ifndef XS_TDM_STAGED
    if (tid < F_) xs[tid] = x[(size_t)b * F_ + tid];
#endif
    __syncthreads();

    // Loop-invariant x values: features k*32 + sl*2, +1 for k = 0..3.
    v2f xk[4];
#pragma unroll
    for (int k = 0; k < 4; ++k)
        xk[k] = *(const v2f*)(xs + k * 32 + sl * 2);   // ds_load_b64, 8B aligned

    const float* repb = rep + (size_t)b * CG_ * (2 * F_);
    const int    cg0  = seg * (CG_ / SEGS);

#pragma unroll
    for (int i = 0; i < (CG_ / SEGS) / (2 * WAVES); ++i) {  // 4 iterations
        const int cg = cg0 + i * (2 * WAVES) + wave * 2 + sub;
        const float* p = repb + (size_t)cg * (2 * F_) + sl * 4;

        float q = 0.0f;
#pragma unroll
        for (int k = 0; k < 4; ++k) {
            // (m0, v0, m1, v1) for features k*32+sl*2, +1. Single-use -> NT.
            v4f a = __builtin_nontemporal_load((const v4f*)(p + k * 64));
            float d0 = xk[k].x - a.x;
            float d1 = xk[k].y - a.z;
            float t0 = d0 * __builtin_amdgcn_rcpf(a.y);   // v_rcp_f32 (TRANS)
            float t1 = d1 * __builtin_amdgcn_rcpf(a.w);
            q = fmaf(t0, t0, fmaf(t1, t1, q));
        }

        // 16-lane xor-butterfly (stays within each half-wave): reduces both
        // outputs' 128-feature sums with 4 shuffle steps total.
        q += __shfl_xor(q, 8, 32);
        q += __shfl_xor(q, 4, 32);
        q += __shfl_xor(q, 2, 32);
        q += __shfl_xor(q, 1, 32);

        if (sl == 0) {  // lanes 0 and 16 store the two outputs in one instr
            __builtin_nontemporal_store(__expf(-0.5f * q),
                                        out + (size_t)b * CG_ + cg);
        }
    }
}

extern "C" void kernel_launch(void* const* d_in, const int* in_sizes, int n_in,
                              void* d_out, int out_size, void* d_ws, size_t ws_size,
                              hipStream_t stream) {
    const float* x   = (const float*)d_in[0];   // [512, 128] f32
    const float* rep = (const float*)d_in[1];   // [512, 64, 16, 128, 2] f32
    float*       out = (float*)d_out;           // [512, 64, 16] f32

    dim3 grid(B_ * SEGS);   // 8192 blocks
    dim3 block(256);        // 8 wave32 per block
    gaussian_quad_kernel<<<grid, block, 0, stream>>>(x, rep, out);
}